// CausalityConstraints_46935402610849
// MI455X (gfx1250) — compile-verified
//
#include <hip/hip_runtime.h>
#include <math.h>

#define BATCH 8
#define SEQ   1024
#define KREL  4
#define NPOS  (BATCH * SEQ)   // 8192

typedef __attribute__((ext_vector_type(2))) float v2f;
typedef __attribute__((ext_vector_type(4))) float v4f;
typedef __attribute__((ext_vector_type(8))) float v8f;

// ---------------------------------------------------------------------------
// Kernel 1: factA4[b,i] = any(rel[b,i,p,k] > 0.5 for p < i) ? 0.7 : 1.0
// (maxrel>0.5 && exists_earlier  ==  exists_earlier).
// Layout [B,S,S,K]: the p<i region is the first 4*i contiguous floats of the
// row -> triangular streaming read with NT hint (don't pollute L2; read-once).
// One block per (b,i) row; 256 threads; <=4 float4 loads per thread.
// ---------------------------------------------------------------------------
__global__ void __launch_bounds__(256)
k_rel_reduce(const float* __restrict__ rel, float* __restrict__ factA4) {
  const int row = blockIdx.x;              // b*SEQ + i
  const int i   = row & (SEQ - 1);
  const v4f* base = (const v4f*)(rel + (size_t)row * (SEQ * KREL)); // 16KB-aligned
  __shared__ int found;
  if (threadIdx.x == 0) found = 0;
  __syncthreads();
  int local = 0;
  for (int j = threadIdx.x; j < i; j += 256) {   // i float4s == 4*i floats (p<i)
    v4f v = __builtin_nontemporal_load(base + j);
    local |= (v.x > 0.5f) || (v.y > 0.5f) || (v.z > 0.5f) || (v.w > 0.5f);
  }
  if (local) atomicOr(&found, 1);
  __syncthreads();
  if (threadIdx.x == 0) factA4[row] = found ? 0.7f : 1.0f;
}

// ---------------------------------------------------------------------------
// Kernel 2: per-position factors.
//   MLP 8->32->16->1 (relu,relu,sigmoid) via V_WMMA_F32_16X16X4_F32.
//   16 positions per wave; 4 waves (128 thr) -> 64 positions per block.
// WMMA f32 layouts (ISA 7.12.2):
//   A 16x4 : lane m=lane&15, vgpr v holds K = v + 2*(lane>>4)   (chunk base k0)
//   B 4x16 : lane n=lane&15, vgpr v holds K = v + 2*(lane>>4)
//   D 16x16: lane n=lane&15, vgpr v holds M = v + 8*(lane>>4)
// LDS round-trip converts D-layout (M across vgprs) into A-layout (M across
// lanes) between layers.
// ---------------------------------------------------------------------------
__global__ void __launch_bounds__(128)
k_mlp_factors(const float* __restrict__ aL, const float* __restrict__ oL,
              const int*   __restrict__ exA, const int* __restrict__ exO,
              const float* __restrict__ W1, const float* __restrict__ b1,
              const float* __restrict__ W2, const float* __restrict__ b2,
              const float* __restrict__ W3, const float* __restrict__ b3,
              float* __restrict__ mult1, float* __restrict__ fA2,
              float* __restrict__ fO2) {
  __shared__ float h[4][16][32];           // per-wave hidden staging
  const int tid  = threadIdx.x;
  const int w    = tid >> 5;               // wave 0..3
  const int lane = tid & 31;
  const int hf   = lane >> 4;              // lane-half
  const int ln   = lane & 15;
  const int posBase = blockIdx.x * 64 + w * 16;
  const int p = posBase + ln;              // position this (ln) column owns

  const float* aRow = aL + (size_t)p * 4;
  const float* oRow = oL + (size_t)p * 4;
  const float x0 = aRow[0], x1 = aRow[1], x2 = aRow[2], x3 = aRow[3];
  const float x4 = oRow[0], x5 = oRow[1], x6 = oRow[2], x7 = oRow[3];

  // ----- layer 1: h1[16x32] = x[16x8] @ W1[8x32], two K-chunks, two N-tiles
  v2f a0, a1, bw;
  a0.x = hf ? x2 : x0;  a0.y = hf ? x3 : x1;     // K-chunk k0=0
  a1.x = hf ? x6 : x4;  a1.y = hf ? x7 : x5;     // K-chunk k0=4
  v8f acc0 = {0,0,0,0,0,0,0,0};
  v8f acc1 = {0,0,0,0,0,0,0,0};
  // N-tile 0
  bw.x = W1[(0 + 2*hf)*32 + ln];  bw.y = W1[(1 + 2*hf)*32 + ln];
  acc0 = __builtin_amdgcn_wmma_f32_16x16x4_f32(false, a0, false, bw, (short)0, acc0, false, false);
  bw.x = W1[(4 + 2*hf)*32 + ln];  bw.y = W1[(5 + 2*hf)*32 + ln];
  acc0 = __builtin_amdgcn_wmma_f32_16x16x4_f32(false, a1, false, bw, (short)0, acc0, false, false);
  // N-tile 1 (channels 16..31)
  bw.x = W1[(0 + 2*hf)*32 + 16 + ln];  bw.y = W1[(1 + 2*hf)*32 + 16 + ln];
  acc1 = __builtin_amdgcn_wmma_f32_16x16x4_f32(false, a0, false, bw, (short)0, acc1, false, false);
  bw.x = W1[(4 + 2*hf)*32 + 16 + ln];  bw.y = W1[(5 + 2*hf)*32 + 16 + ln];
  acc1 = __builtin_amdgcn_wmma_f32_16x16x4_f32(false, a1, false, bw, (short)0, acc1, false, false);

  const float bia0 = b1[ln], bia1 = b1[16 + ln];
  #pragma unroll
  for (int v = 0; v < 8; ++v) {            // bias + relu, D-layout -> LDS [pos][ch]
    float t0 = acc0[v] + bia0;  h[w][v + 8*hf][ln]      = t0 > 0.0f ? t0 : 0.0f;
    float t1 = acc1[v] + bia1;  h[w][v + 8*hf][16 + ln] = t1 > 0.0f ? t1 : 0.0f;
  }
  __syncthreads();

  // ----- layer 2: h2[16x16] = h1[16x32] @ W2[32x16], 8 accumulating K-chunks
  v8f acc2 = {0,0,0,0,0,0,0,0};
  #pragma unroll
  for (int k0 = 0; k0 < 32; k0 += 4) {
    v2f aa, bb;
    aa.x = h[w][ln][k0 + 2*hf];
    aa.y = h[w][ln][k0 + 1 + 2*hf];
    bb.x = W2[(k0 + 2*hf)*16 + ln];
    bb.y = W2[(k0 + 1 + 2*hf)*16 + ln];
    acc2 = __builtin_amdgcn_wmma_f32_16x16x4_f32(false, aa, false, bb, (short)0, acc2, false, false);
  }
  __syncthreads();
  const float bia2 = b2[ln];
  #pragma unroll
  for (int v = 0; v < 8; ++v) {
    float t = acc2[v] + bia2;
    h[w][v + 8*hf][ln] = t > 0.0f ? t : 0.0f;   // h2 -> LDS [pos][ch]
  }
  __syncthreads();

  // ----- layer 3 + per-position factors: lanes 0..15 handle their position
  if (hf == 0) {
    float s3 = b3[0];
    #pragma unroll
    for (int c = 0; c < 16; ++c) s3 += h[w][ln][c] * W3[c];
    float cg   = 1.0f / (1.0f + expf(-s3));             // sigmoid
    float mult = cg < 0.5f ? 2.0f * cg : 1.0f;
    mult1[p] = mult;

    // softmax heads: implicit prob = (e0+e1)/sum
    float mA = fmaxf(fmaxf(x0, x1), fmaxf(x2, x3));
    float e0 = expf(x0 - mA), e1 = expf(x1 - mA), e2 = expf(x2 - mA), e3 = expf(x3 - mA);
    float impA = (e0 + e1) / (e0 + e1 + e2 + e3);
    float mO = fmaxf(fmaxf(x4, x5), fmaxf(x6, x7));
    float g0 = expf(x4 - mO), g1 = expf(x5 - mO), g2 = expf(x6 - mO), g3 = expf(x7 - mO);
    float impO = (g0 + g1) / (g0 + g1 + g2 + g3);

    const int bI = p >> 10;
    const int s  = p & (SEQ - 1);
    const int* eA = exA + bI * SEQ;
    const int* eO = exO + bI * SEQ;
    bool nearA = false, nearO = false;
    #pragma unroll
    for (int d = -3; d <= 3; ++d) {
      int ss = s + d;
      if (ss >= 0 && ss < SEQ) { nearA |= (eA[ss] > 0); nearO |= (eO[ss] > 0); }
    }
    fA2[p] = (impA > 0.5f && !nearO) ? 0.3f : 1.0f;
    fO2[p] = (impO > 0.5f && !nearA) ? 0.3f : 1.0f;
  }
}

// ---------------------------------------------------------------------------
// Kernel 3: the lax.scan. Carry state is only newAct[i-1], newAct[i-2]; we
// precompute 3 bitmaps per position (origAct, act-if-fact3=1.0, act-if=0.1),
// run a 2-bit boolean recurrence serially (thread 0), then write rows in
// parallel. One block per batch.
// ---------------------------------------------------------------------------
__global__ void __launch_bounds__(256)
k_scan_write(const float* __restrict__ aL, const float* __restrict__ oL,
             const float* __restrict__ mult1, const float* __restrict__ fA2,
             const float* __restrict__ fO2,  const float* __restrict__ fA4,
             float* __restrict__ outA, float* __restrict__ outO) {
  const int b = blockIdx.x;
  const int t = threadIdx.x;
  __shared__ float sm[SEQ], sfa[SEQ], sfo[SEQ];
  __shared__ unsigned OG[32], IF1[32], IF01[32], F3[32];
  if (t < 32) { OG[t] = 0u; IF1[t] = 0u; IF01[t] = 0u; }
  __syncthreads();

  // Phase A: per-position prework + activity bitmaps
  for (int s = t; s < SEQ; s += 256) {
    const int p = b * SEQ + s;
    v4f av = *(const v4f*)(aL + (size_t)p * 4);
    v4f ov = *(const v4f*)(oL + (size_t)p * 4);
    float mult = mult1[p];
    float fa1  = fA2[p] * fA4[p];   // fact3-independent part for aspects
    float fo1  = fO2[p];            // for opinions
    sm[s] = mult;  sfa[s] = fa1;  sfo[s] = fo1;

    float a0 = av.x * mult, a1 = av.y * mult, a2 = av.z * mult, a3 = av.w * mult;
    float o0 = ov.x * mult, o1 = ov.y * mult, o2 = ov.z * mult, o3 = ov.w * mult;
    bool orig = (fmaxf(fmaxf(av.x, av.y), fmaxf(av.z, av.w)) > 0.5f) ||
                (fmaxf(fmaxf(ov.x, ov.y), fmaxf(ov.z, ov.w)) > 0.5f);
    // activity of the final (written) row for fact3 in {1.0, 0.1}
    float fa = fa1, fo = fo1;
    bool act1  = (fmaxf(fmaxf(a0*fa, a1*fa), fmaxf(a2, a3)) > 0.5f) ||
                 (fmaxf(fmaxf(o0*fo, o1*fo), fmaxf(o2, o3)) > 0.5f);
    fa = fa1 * 0.1f; fo = fo1 * 0.1f;
    bool act01 = (fmaxf(fmaxf(a0*fa, a1*fa), fmaxf(a2, a3)) > 0.5f) ||
                 (fmaxf(fmaxf(o0*fo, o1*fo), fmaxf(o2, o3)) > 0.5f);

    unsigned bit = 1u << (s & 31);
    if (orig)  atomicOr(&OG[s >> 5],  bit);
    if (act1)  atomicOr(&IF1[s >> 5], bit);
    if (act01) atomicOr(&IF01[s >> 5], bit);
  }
  __syncthreads();

  // Phase B: serial 2-bit recurrence over 1024 positions (thread 0)
  if (t == 0) {
    unsigned n1 = 0u, n2 = 0u;     // newAct[s-1], newAct[s-2]
    for (int wi = 0; wi < 32; ++wi) {
      unsigned og   = OG[wi];
      unsigned ogn  = (wi < 31) ? OG[wi + 1] : 0u;
      unsigned if1  = IF1[wi];
      unsigned if01 = IF01[wi];
      unsigned f3w  = 0u;
      #pragma unroll 4
      for (int bb = 0; bb < 32; ++bb) {
        unsigned o1b = (bb + 1 < 32) ? ((og >> (bb + 1)) & 1u) : (ogn & 1u);
        unsigned o2b = (bb + 2 < 32) ? ((og >> (bb + 2)) & 1u) : ((ogn >> (bb + 2 - 32)) & 1u);
        unsigned nb  = n1 | n2 | o1b | o2b;       // neighbor active -> fact3=1.0
        unsigned na  = nb ? ((if1 >> bb) & 1u) : ((if01 >> bb) & 1u);
        f3w |= nb << bb;
        n2 = n1;  n1 = na;
      }
      F3[wi] = f3w;
    }
  }
  __syncthreads();

  // Phase C: parallel final writes
  for (int s = t; s < SEQ; s += 256) {
    const int p = b * SEQ + s;
    v4f av = *(const v4f*)(aL + (size_t)p * 4);
    v4f ov = *(const v4f*)(oL + (size_t)p * 4);
    float mult = sm[s];
    float f3 = ((F3[s >> 5] >> (s & 31)) & 1u) ? 1.0f : 0.1f;
    float fa = sfa[s] * f3;
    float fo = sfo[s] * f3;
    v4f ra, ro;
    ra.x = av.x * mult * fa;  ra.y = av.y * mult * fa;
    ra.z = av.z * mult;       ra.w = av.w * mult;
    ro.x = ov.x * mult * fo;  ro.y = ov.y * mult * fo;
    ro.z = ov.z * mult;       ro.w = ov.w * mult;
    *(v4f*)(outA + (size_t)p * 4) = ra;
    *(v4f*)(outO + (size_t)p * 4) = ro;
  }
}

// ---------------------------------------------------------------------------
extern "C" void kernel_launch(void* const* d_in, const int* in_sizes, int n_in,
                              void* d_out, int out_size, void* d_ws, size_t ws_size,
                              hipStream_t stream) {
  (void)in_sizes; (void)n_in; (void)out_size; (void)ws_size;
  const float* aL  = (const float*)d_in[0];
  const float* oL  = (const float*)d_in[1];
  const float* rel = (const float*)d_in[2];
  const int*   exA = (const int*)d_in[3];
  const int*   exO = (const int*)d_in[4];
  const float* W1  = (const float*)d_in[5];
  const float* b1  = (const float*)d_in[6];
  const float* W2  = (const float*)d_in[7];
  const float* b2  = (const float*)d_in[8];
  const float* W3  = (const float*)d_in[9];
  const float* b3  = (const float*)d_in[10];

  float* ws    = (float*)d_ws;              // 4 * 8192 floats = 128 KB scratch
  float* fA4   = ws;
  float* mult1 = ws + NPOS;
  float* fA2   = ws + 2 * NPOS;
  float* fO2   = ws + 3 * NPOS;
  float* outA  = (float*)d_out;
  float* outO  = outA + NPOS * 4;

  k_rel_reduce<<<NPOS, 256, 0, stream>>>(rel, fA4);
  k_mlp_factors<<<NPOS / 64, 128, 0, stream>>>(aL, oL, exA, exO,
                                               W1, b1, W2, b2, W3, b3,
                                               mult1, fA2, fO2);
  k_scan_write<<<BATCH, 256, 0, stream>>>(aL, oL, mult1, fA2, fO2, fA4,
                                          outA, outO);
}